// ContrastiveLoss_55920474194486
// MI455X (gfx1250) — compile-verified
//
#include <hip/hip_runtime.h>

typedef __attribute__((ext_vector_type(2))) float v2f;
typedef __attribute__((ext_vector_type(8))) float v8f;

#define NB 64
#define NR 40
#define NW 48
#define ND 512
#define LAM_SM 9.0f
#define LAM_LSE 6.0f
#define MARGIN_ 0.2f
#define EPS_ 1e-8f

#define WPB 8           // waves per block (one (c,b) pair per wave)
#define RSTRIDE 42      // LDS r-stride (floats) for transposed 48(w) x 40(r) tile
                        // even -> 8B-aligned v2f loads; gcd(42,64)=2 -> near conflict-free

__global__ __launch_bounds__(256) void scores_kernel(
    const float* __restrict__ im, const float* __restrict__ s,
    const int* __restrict__ s_l, const int* __restrict__ k_part,
    float* __restrict__ scores)
{
    __shared__ float attnLDS[WPB][NW * RSTRIDE];   // At[w*RSTRIDE + r]
    __shared__ float exLDS[WPB][NW];

    const int lane = threadIdx.x & 31;
    const int wave = threadIdx.x >> 5;
    const int l16  = lane & 15;
    const int hh   = lane >> 4;     // lane half: 0 or 1
    const int koff = hh * 2;        // A/B frag K offset per half

    const int pair = blockIdx.x * WPB + wave;   // 4096 pairs
    const int b = pair >> 6;
    const int c = pair & 63;

    const float* imb = im + (size_t)b * (NR * ND);
    const float* sc  = s  + (size_t)c * (NW * ND);
    const int sl = s_l[c];
    const int kp = k_part[b];

    float* At = attnLDS[wave];

    // ---------- GEMM1: attn(40x48) = im[b](40x512) @ s[c]^T ----------
    // A rows 40..47 are clamped to row 0 (in-bounds, finite); the resulting
    // C rows 40..47 are garbage but never read downstream -> no masking needed.
    v8f acc[3][3] = {};

    const v2f* aP[3];
#pragma unroll
    for (int mt = 0; mt < 3; ++mt) {
        int m = mt * 16 + l16;
        int mc = (m < NR) ? m : 0;
        aP[mt] = (const v2f*)(imb + mc * ND + koff);
    }
    const v2f* bP[3];
#pragma unroll
    for (int nt = 0; nt < 3; ++nt) {
        int w = nt * 16 + l16;
        bP[nt] = (const v2f*)(sc + w * ND + koff);
    }

    for (int k = 0; k < ND; k += 4) {
        int k2 = k >> 1;    // index in v2f units
        v2f aF[3], bF[3];
#pragma unroll
        for (int mt = 0; mt < 3; ++mt) aF[mt] = aP[mt][k2];
#pragma unroll
        for (int nt = 0; nt < 3; ++nt) bF[nt] = bP[nt][k2];
#pragma unroll
        for (int mt = 0; mt < 3; ++mt)
#pragma unroll
            for (int nt = 0; nt < 3; ++nt)
                acc[mt][nt] = __builtin_amdgcn_wmma_f32_16x16x4_f32(
                    false, aF[mt], false, bF[nt], (short)0, acc[mt][nt],
                    false, false);
    }

    // ---------- LeakyReLU(0.1) + word mask, spill transposed: At[w][r] ----------
#pragma unroll
    for (int mt = 0; mt < 3; ++mt) {
#pragma unroll
        for (int nt = 0; nt < 3; ++nt) {
            int w = nt * 16 + l16;
            float wm = (w < sl) ? 1.0f : 0.0f;
#pragma unroll
            for (int v = 0; v < 8; ++v) {
                int r = mt * 16 + v + hh * 8;   // C/D layout: M = v + 8*half
                if (r < NR) {
                    float x = acc[mt][nt][v];
                    x = (x > 0.0f) ? x : 0.1f * x;
                    At[w * RSTRIDE + r] = x * wm;
                }
            }
        }
    }
    __syncthreads();

    // ---------- per-region (r) L2 normalize over words w ----------
    for (int rr = lane; rr < NR; rr += 32) {
        float ssum = 0.0f;
#pragma unroll 8
        for (int w = 0; w < NW; ++w) { float x = At[w * RSTRIDE + rr]; ssum += x * x; }
        float inv = 1.0f / (sqrtf(ssum) + EPS_);
#pragma unroll 8
        for (int w = 0; w < NW; ++w) At[w * RSTRIDE + rr] *= inv;
    }
    __syncthreads();

    // ---------- per-word exp(lambda*attn) with region mask ----------
    // (softmax denominator cancels in the cosine similarity -> skipped)
    for (int w = lane; w < NW; w += 32) {
        for (int r = 0; r < NR; ++r) {
            float x = At[w * RSTRIDE + r];
            At[w * RSTRIDE + r] = (r < kp) ? __expf(LAM_SM * x) : 0.0f;
        }
    }
    __syncthreads();

    // ---------- GEMM2 + fused epilogue ----------
    // wc(48x512) = E(48x40) @ im[b](40x512), streamed in 16-wide D chunks;
    // accumulate num = s.wc, nrm = |wc|^2, s2 = |s|^2 per lane.
    v2f aF2[10][3];     // A-frags hoisted out of the D loop (b64 LDS loads)
#pragma unroll
    for (int kk = 0; kk < 10; ++kk) {
        int r0 = kk * 4 + koff;
#pragma unroll
        for (int mt = 0; mt < 3; ++mt) {
            int w = mt * 16 + l16;
            aF2[kk][mt] = *(const v2f*)(At + w * RSTRIDE + r0);
        }
    }

    v8f numA[3] = {}, nrmA[3] = {}, s2A[3] = {};

    for (int d0 = 0; d0 < ND; d0 += 16) {
        v8f t[3] = {};
#pragma unroll
        for (int kk = 0; kk < 10; ++kk) {
            int r0 = kk * 4 + koff;
            v2f bF;
            bF.x = imb[r0 * ND + d0 + l16];
            bF.y = imb[(r0 + 1) * ND + d0 + l16];
#pragma unroll
            for (int mt = 0; mt < 3; ++mt)
                t[mt] = __builtin_amdgcn_wmma_f32_16x16x4_f32(
                    false, aF2[kk][mt], false, bF, (short)0, t[mt],
                    false, false);
        }
#pragma unroll
        for (int mt = 0; mt < 3; ++mt) {
#pragma unroll
            for (int v = 0; v < 8; ++v) {
                int w = mt * 16 + v + hh * 8;
                float sv = sc[w * ND + d0 + l16];
                float tv = t[mt][v];
                numA[mt][v] += tv * sv;
                nrmA[mt][v] += tv * tv;
                s2A[mt][v]  += sv * sv;
            }
        }
    }

    // ---------- reduce over the 16 lanes holding each row w ----------
    float* ex = exLDS[wave];
#pragma unroll
    for (int mt = 0; mt < 3; ++mt) {
#pragma unroll
        for (int v = 0; v < 8; ++v) {
            float n1 = numA[mt][v], n2 = nrmA[mt][v], n3 = s2A[mt][v];
#pragma unroll
            for (int off = 1; off < 16; off <<= 1) {
                n1 += __shfl_xor(n1, off, 32);
                n2 += __shfl_xor(n2, off, 32);
                n3 += __shfl_xor(n3, off, 32);
            }
            if (l16 == 0) {
                int w = mt * 16 + v + hh * 8;
                float den = fmaxf(sqrtf(n2 * n3), EPS_);
                float rim = n1 / den;                 // cosine similarity
                ex[w] = (w < sl) ? __expf(LAM_LSE * rim) : 0.0f;
            }
        }
    }
    __syncthreads();

    if (lane == 0) {
        float sumv = 0.0f;
#pragma unroll 8
        for (int w = 0; w < NW; ++w) sumv += ex[w];
        scores[b * NB + c] = __logf(sumv) / LAM_LSE;  // scores[image][caption]
    }
}

// ---------------- loss from the 64x64 score matrix ----------------
__global__ __launch_bounds__(64) void loss_kernel(const float* __restrict__ scores,
                                                  float* __restrict__ out)
{
    __shared__ float diag[NB];
    __shared__ float redSemi[NB];
    __shared__ float redFull[NB];
    int i = threadIdx.x;
    diag[i] = scores[i * NB + i];
    __syncthreads();

    float di = diag[i];
    float cs = 0.0f, scs = 0.0f, cim = 0.0f, scim = 0.0f;
    for (int j = 0; j < NB; ++j) {
        if (j == i) continue;
        // row pass (cost_s: d1 = diag of row i)
        float a = scores[i * NB + j];
        float v = fmaxf(MARGIN_ + a - di, 0.0f);
        cs = fmaxf(cs, v);
        if (!(a > di)) scs = fmaxf(scs, v);
        // column pass (cost_im: column i, d2 = diag[i])
        float bsc = scores[j * NB + i];
        float v2 = fmaxf(MARGIN_ + bsc - di, 0.0f);
        cim = fmaxf(cim, v2);
        if (!(bsc > di)) scim = fmaxf(scim, v2);
    }
    redSemi[i] = scs + scim;
    redFull[i] = cs + cim;
    __syncthreads();
    if (i == 0) {
        float sa = 0.0f, sb = 0.0f;
        for (int t = 0; t < NB; ++t) { sa += redSemi[t]; sb += redFull[t]; }
        out[0] = sa / (float)NB;
        out[1] = sb / (float)NB;
    }
}

extern "C" void kernel_launch(void* const* d_in, const int* in_sizes, int n_in,
                              void* d_out, int out_size, void* d_ws, size_t ws_size,
                              hipStream_t stream) {
    const float* im     = (const float*)d_in[0];
    const float* s      = (const float*)d_in[1];
    const int*   s_l    = (const int*)d_in[2];
    const int*   k_part = (const int*)d_in[3];
    float* scores = (float*)d_ws;               // 64*64 floats of scratch
    float* out    = (float*)d_out;              // 2 floats

    scores_kernel<<<(NB * NB) / WPB, 32 * WPB, 0, stream>>>(im, s, s_l, k_part, scores);
    loss_kernel<<<1, NB, 0, stream>>>(scores, out);
}